// FragTreeMPNN_19713899888723
// MI455X (gfx1250) — compile-verified
//
#include <hip/hip_runtime.h>
#include <hip/hip_bf16.h>
#include <math.h>

// ---------------- problem constants (mirror reference module) ----------------
constexpr int ATOM_DIM  = 21;
constexpr int BOND_DIM  = 9;
constexpr int HID       = 128;
constexpr int N_LAYERS  = 3;
constexpr int MAX_DEPTH = 8;
constexpr int N_ATOMS   = 200000;
constexpr int N_EDGES   = 400000;
constexpr int N_BONDS   = 400000;
constexpr int N_FRAGS   = 500000;
constexpr int MZ        = 1000;
constexpr int REP_DIM   = 2 * HID + BOND_DIM;  // 265
constexpr int KPAD      = 268;                 // 265 padded up to multiple of 4
constexpr int RT        = 10;                  // row tiles per GEMM block (12500 % 10 == 0)
constexpr int NT        = 4;                   // bond tiles per bond-MLP block (64 bonds)
constexpr int APAD      = HID + 4;             // LDS row stride to kill bank conflicts
constexpr int HPAD      = HID + 1;             // Hd row stride (1 row per bank)

// ---------------- WMMA fp32 16x16x4 helper ----------------
typedef __attribute__((ext_vector_type(2))) float v2f;
typedef __attribute__((ext_vector_type(8))) float v8f;

__device__ __forceinline__ v8f wmma_f32_k4(v2f a, v2f b, v8f c) {
  // D(16x16,f32) = A(16x4,f32) * B(4x16,f32) + C
  return __builtin_amdgcn_wmma_f32_16x16x4_f32(
      /*neg_a=*/false, a, /*neg_b=*/false, b,
      /*c_mod=*/(short)0, c, /*reuse_a=*/false, /*reuse_b=*/false);
}

// order-preserving float<->uint for atomicMax-based segment max
__device__ __forceinline__ unsigned fkey(float f) {
  unsigned u = __float_as_uint(f);
  return (u & 0x80000000u) ? ~u : (u | 0x80000000u);
}
__device__ __forceinline__ float fdec(unsigned k) {
  return (k & 0x80000000u) ? __uint_as_float(k & 0x7fffffffu)
                           : __uint_as_float(~k);
}

// ---------------- kernels ----------------

// h = relu(x @ W_in + b_in) ; one block (128 thr) per atom
__global__ __launch_bounds__(HID) void k_input(const float* __restrict__ x,
                                               const float* __restrict__ W_in,
                                               const float* __restrict__ b_in,
                                               float* __restrict__ h) {
  __shared__ float xr[ATOM_DIM];
  const int atom = blockIdx.x;
  const int tid = threadIdx.x;
  if (tid < ATOM_DIM) xr[tid] = x[atom * ATOM_DIM + tid];
  __syncthreads();
  float acc = b_in[tid];
  #pragma unroll
  for (int k = 0; k < ATOM_DIM; ++k) acc += xr[k] * W_in[k * HID + tid];
  h[(size_t)atom * HID + tid] = acc > 0.f ? acc : 0.f;
}

__global__ void k_fill(float* __restrict__ p, float v, int n) {
  int i = blockIdx.x * blockDim.x + threadIdx.x;
  if (i < n) p[i] = v;
}

__global__ void k_deg_scatter(float* __restrict__ deg,
                              const int* __restrict__ dst, int nedges) {
  int e = blockIdx.x * blockDim.x + threadIdx.x;
  if (e < nedges) atomicAdd(&deg[dst[e]], 1.0f);
}

__global__ void k_rsqrt_inplace(float* __restrict__ p, int n) {
  int i = blockIdx.x * blockDim.x + threadIdx.x;
  if (i < n) p[i] = rsqrtf(p[i]);
}

// hw = h @ W (nx128 @ 128x128), agg = hw * dinv^2 (self term), via fp32 WMMA.
// Block: 256 thr = 8 waves; wave w owns cols [16w,16w+16) and caches its
// 32 B-fragments (64 VGPRs) once, then sweeps RT row tiles of 16 rows.
// A rows padded to APAD floats so the 16 lanes of a half-wave hit 16 banks.
__global__ __launch_bounds__(256) void k_gemm_hw(const float* __restrict__ h,
                                                 const float* __restrict__ W,
                                                 const float* __restrict__ dinv,
                                                 float* __restrict__ hw,
                                                 float* __restrict__ agg) {
  __shared__ __align__(16) float A[16][APAD];
  const int tid = threadIdx.x;
  const int wave = tid >> 5, lane = tid & 31;
  const int r = lane & 15;
  const int koff = (lane < 16) ? 0 : 2;
  const int col = wave * 16 + r;

  // cache B fragments for this wave's 16-column slice of W
  v2f B[32];
  #pragma unroll
  for (int kk = 0; kk < 32; ++kk) {
    const int ka = kk * 4 + koff;
    B[kk].x = W[ka * HID + col];
    B[kk].y = W[(ka + 1) * HID + col];
  }

  for (int t = 0; t < RT; ++t) {
    const int rowBase = (blockIdx.x * RT + t) * 16;
    __syncthreads();  // protect A from previous iteration's readers
    for (int i = tid; i < 16 * (HID / 4); i += 256) {
      const int rr = i >> 5, q = i & 31;  // 32 float4 per row
      *(float4*)&A[rr][q * 4] =
          ((const float4*)(h + (size_t)(rowBase + rr) * HID))[q];
    }
    __syncthreads();
    v8f c = {};
    #pragma unroll
    for (int kk = 0; kk < 32; ++kk) {
      const int ka = kk * 4 + koff;
      v2f a; a.x = A[r][ka]; a.y = A[r][ka + 1];
      c = wmma_f32_k4(a, B[kk], c);
    }
    #pragma unroll
    for (int j = 0; j < 8; ++j) {
      const int row = rowBase + j + ((lane < 16) ? 0 : 8);
      const float v = c[j];
      hw[(size_t)row * HID + col] = v;
      const float dv = dinv[row];
      agg[(size_t)row * HID + col] = v * dv * dv;
    }
  }
}

// agg[dst] += hw[src] * dinv[src]*dinv[dst] ; one thread per (edge, 4 channels)
__global__ void k_edge_scatter(float* __restrict__ agg,
                               const float* __restrict__ hw,
                               const int* __restrict__ src,
                               const int* __restrict__ dst,
                               const float* __restrict__ dinv, int nedges) {
  const long long idx = (long long)blockIdx.x * blockDim.x + threadIdx.x;
  if (idx >= (long long)nedges * (HID / 4)) return;
  const int e = (int)(idx >> 5);
  const int q = (int)(idx & 31);
  const int s = src[e], d = dst[e];
  const float nrm = dinv[s] * dinv[d];
  const float4 v = ((const float4*)(hw + (size_t)s * HID))[q];
  float* base = agg + (size_t)d * HID + q * 4;
  atomicAdd(base + 0, v.x * nrm);
  atomicAdd(base + 1, v.y * nrm);
  atomicAdd(base + 2, v.z * nrm);
  atomicAdd(base + 3, v.w * nrm);
}

// h = relu(agg + b) (vectorized x4)
__global__ void k_bias_relu(float* __restrict__ h, const float* __restrict__ agg,
                            const float* __restrict__ b, int n4) {
  int i = blockIdx.x * blockDim.x + threadIdx.x;
  if (i < n4) {
    const int c = (i & 31) * 4;
    float4 v = ((const float4*)agg)[i];
    v.x += b[c + 0]; v.y += b[c + 1]; v.z += b[c + 2]; v.w += b[c + 3];
    v.x = v.x > 0.f ? v.x : 0.f;
    v.y = v.y > 0.f ? v.y : 0.f;
    v.z = v.z > 0.f ? v.z : 0.f;
    v.w = v.w > 0.f ? v.w : 0.f;
    ((float4*)h)[i] = v;
  }
}

// zero-pad W1 (265x128) -> W1p (268x128)
__global__ void k_w1pad(const float* __restrict__ W1, float* __restrict__ W1p) {
  int i = blockIdx.x * blockDim.x + threadIdx.x;
  if (i < KPAD * HID) {
    int row = i / HID, col = i % HID;
    W1p[i] = (row < REP_DIM) ? W1[row * HID + col] : 0.f;
  }
}

// bond MLP: reps=[h[i]||h[j]||ef] (pad->268) @ W1p -> relu -> @ W2 + b2
// block = 256 thr, NT*16 = 64 bonds; each wave owns a 16-col slice of hidden
// and keeps NT accumulators so every B-fragment load feeds NT WMMAs.
__global__ __launch_bounds__(256) void k_bond_mlp(
    const float* __restrict__ h, const int* __restrict__ atom_i,
    const int* __restrict__ atom_j, const float* __restrict__ ef,
    const float* __restrict__ W1p, const float* __restrict__ b1,
    const float* __restrict__ W2, const float* __restrict__ b2,
    float* __restrict__ scores, float* __restrict__ cs) {
  __shared__ __align__(16) float R[NT][16][KPAD];   // ~68.6 KB
  __shared__ float Hd[NT * 16][HPAD];               // ~33 KB
  const int bondBase = blockIdx.x * (16 * NT);
  const int tid = threadIdx.x;
  // cols [0,256): vectorized gather of h[atom_i] and h[atom_j]
  for (int i = tid; i < NT * 16 * 64; i += 256) {
    const int b = i >> 6, q = i & 63;        // q: which float4 of 64
    const int bond = bondBase + b;
    const float* srcp = (q < 32)
        ? (h + (size_t)atom_i[bond] * HID + q * 4)
        : (h + (size_t)atom_j[bond] * HID + (q - 32) * 4);
    *(float4*)&R[b >> 4][b & 15][q * 4] = *(const float4*)srcp;
  }
  // cols [256,268): edge features + zero pad
  for (int i = tid; i < NT * 16 * 12; i += 256) {
    const int b = i / 12, k = 256 + (i % 12);
    R[b >> 4][b & 15][k] =
        (k < REP_DIM) ? ef[(size_t)(bondBase + b) * BOND_DIM + (k - 256)] : 0.f;
  }
  __syncthreads();
  const int wave = tid >> 5, lane = tid & 31;
  const int r = lane & 15;
  const int koff = (lane < 16) ? 0 : 2;
  const int col = wave * 16 + r;
  v8f c[NT];
  #pragma unroll
  for (int t = 0; t < NT; ++t) c[t] = (v8f){};
  for (int k0 = 0; k0 < KPAD; k0 += 4) {
    const int ka = k0 + koff;
    v2f b; b.x = W1p[ka * HID + col]; b.y = W1p[(ka + 1) * HID + col];
    #pragma unroll
    for (int t = 0; t < NT; ++t) {
      v2f a; a.x = R[t][r][ka]; a.y = R[t][r][ka + 1];
      c[t] = wmma_f32_k4(a, b, c[t]);
    }
  }
  const float bias = b1[col];
  #pragma unroll
  for (int t = 0; t < NT; ++t) {
    #pragma unroll
    for (int j = 0; j < 8; ++j) {
      const int row = t * 16 + j + ((lane < 16) ? 0 : 8);
      const float v = c[t][j] + bias;
      Hd[row][col] = v > 0.f ? v : 0.f;
    }
  }
  __syncthreads();
  if (tid < NT * 16 * 2) {  // 128 threads: 64 bonds x 2 outputs
    const int rb = tid >> 1, j = tid & 1;
    float acc = b2[j];
    for (int k = 0; k < HID; ++k) acc += Hd[rb][k] * W2[k * 2 + j];
    if (j == 0) scores[bondBase + rb] = acc;
    else        cs[bondBase + rb] = 1.f / (1.f + expf(-acc));
  }
}

__global__ void k_seg_init(unsigned* __restrict__ key, float* __restrict__ den,
                           int n) {
  int i = blockIdx.x * blockDim.x + threadIdx.x;
  if (i < n) { key[i] = fkey(-INFINITY); den[i] = 0.f; }
}

__global__ void k_seg_max(unsigned* __restrict__ key,
                          const float* __restrict__ scores,
                          const int* __restrict__ gid, int n) {
  int b = blockIdx.x * blockDim.x + threadIdx.x;
  if (b < n) atomicMax(&key[gid[b]], fkey(scores[b]));
}

__global__ void k_exp(float* __restrict__ e, float* __restrict__ den,
                      const unsigned* __restrict__ key,
                      const float* __restrict__ scores,
                      const int* __restrict__ gid, int n) {
  int b = blockIdx.x * blockDim.x + threadIdx.x;
  if (b < n) {
    const int g = gid[b];
    const float ev = expf(scores[b] - fdec(key[g]));
    e[b] = ev;
    atomicAdd(&den[g], ev);
  }
}

__global__ void k_lp(float* __restrict__ lp, const float* __restrict__ e,
                     const float* __restrict__ den,
                     const int* __restrict__ gid, int n) {
  int b = blockIdx.x * blockDim.x + threadIdx.x;
  if (b < n) lp[b] = e[b] / (den[gid[b]] + 1e-16f);
}

__global__ void k_ex_init(float* __restrict__ ex,
                          const int* __restrict__ root, int n) {
  int i = blockIdx.x * blockDim.x + threadIdx.x;
  if (i < n) ex[i] = (i == root[0]) ? 1.0f : 0.0f;
}

// wbuf[b] = (depth==d [&& child_b>=0 if phase B]) ? ex[parent]*lp*(cs or 1-cs) : 0
__global__ void k_w_compute(float* __restrict__ w, const float* __restrict__ ex,
                            const int* __restrict__ parent,
                            const float* __restrict__ lp,
                            const float* __restrict__ cs,
                            const int* __restrict__ depth,
                            const int* __restrict__ child_b, int d, int isB,
                            int n) {
  int b = blockIdx.x * blockDim.x + threadIdx.x;
  if (b >= n) return;
  bool m = (depth[b] == d);
  if (isB) m = m && (child_b[b] >= 0);
  float v = 0.f;
  if (m) {
    const float p = cs[b];
    v = ex[parent[b]] * lp[b] * (isB ? (1.f - p) : p);
  }
  w[b] = v;
}

__global__ void k_w_scatter(float* __restrict__ ex,
                            const int* __restrict__ child,
                            const float* __restrict__ w,
                            const int* __restrict__ depth,
                            const int* __restrict__ child_b, int d, int isB,
                            int n) {
  int b = blockIdx.x * blockDim.x + threadIdx.x;
  if (b >= n) return;
  bool m = (depth[b] == d);
  if (isB) m = m && (child_b[b] >= 0);
  if (m) {
    int c = child[b]; if (c < 0) c = 0;
    atomicAdd(&ex[c], w[b]);
  }
}

__global__ void k_spectrum(float* __restrict__ spec,
                           const float* __restrict__ ex,
                           const int* __restrict__ child_a,
                           const int* __restrict__ child_b,
                           const int* __restrict__ ma,
                           const int* __restrict__ mb, int n) {
  int b = blockIdx.x * blockDim.x + threadIdx.x;
  if (b >= n) return;
  int ca = child_a[b]; if (ca < 0) ca = 0;
  int cb = child_b[b]; int cbs = cb < 0 ? 0 : cb;
  if (ma[b] >= 0) atomicAdd(&spec[ma[b]], ex[ca]);
  if (mb[b] >= 0 && cb >= 0) atomicAdd(&spec[mb[b]], ex[cbs]);
}

// ---------------- host launcher ----------------
extern "C" void kernel_launch(void* const* d_in, const int* in_sizes, int n_in,
                              void* d_out, int out_size, void* d_ws, size_t ws_size,
                              hipStream_t stream) {
  const float* x          = (const float*)d_in[0];
  const int*   edge_index = (const int*)d_in[1];   // [2, N_EDGES]
  const int*   atom_i     = (const int*)d_in[2];
  const int*   atom_j     = (const int*)d_in[3];
  const float* edge_feat  = (const float*)d_in[4];
  const int*   group_id   = (const int*)d_in[5];
  const int*   parent     = (const int*)d_in[6];
  const int*   child_a    = (const int*)d_in[7];
  const int*   child_b    = (const int*)d_in[8];
  const int*   mass_a     = (const int*)d_in[9];
  const int*   mass_b     = (const int*)d_in[10];
  const int*   depth      = (const int*)d_in[11];
  const int*   root_idx   = (const int*)d_in[13];  // device scalar
  const float* W_in       = (const float*)d_in[15];
  const float* b_in       = (const float*)d_in[16];
  const float* W_convs    = (const float*)d_in[17];
  const float* b_convs    = (const float*)d_in[18];
  const float* W1         = (const float*)d_in[19];
  const float* b1         = (const float*)d_in[20];
  const float* W2         = (const float*)d_in[21];
  const float* b2         = (const float*)d_in[22];
  float* spec = (float*)d_out;

  const int* src = edge_index;
  const int* dst = edge_index + N_EDGES;

  // ---- workspace carve ----
  float* ws = (float*)d_ws;
  float* h      = ws;                         ws += (size_t)N_ATOMS * HID;
  float* hw     = ws;                         ws += (size_t)N_ATOMS * HID;
  float* agg    = ws;                         ws += (size_t)N_ATOMS * HID;
  float* dinv   = ws;                         ws += N_ATOMS;       // deg then dinv
  float* W1p    = ws;                         ws += (size_t)KPAD * HID;
  float* scores = ws;                         ws += N_BONDS;
  float* cs     = ws;                         ws += N_BONDS;
  float* eArr   = ws;                         ws += N_BONDS;
  float* lp     = ws;                         ws += N_BONDS;
  float* wbuf   = ws;                         ws += N_BONDS;
  unsigned* segkey = (unsigned*)ws;           ws += N_FRAGS;
  float* denom  = ws;                         ws += N_FRAGS;
  float* ex     = ws;                         ws += N_FRAGS;

  const int T = 256;
  auto cdiv = [](long long a, long long b) { return (int)((a + b - 1) / b); };

  // pad W1
  k_w1pad<<<cdiv(KPAD * HID, T), T, 0, stream>>>(W1, W1p);

  // input layer
  k_input<<<N_ATOMS, HID, 0, stream>>>(x, W_in, b_in, h);

  // degree -> dinv
  k_fill<<<cdiv(N_ATOMS, T), T, 0, stream>>>(dinv, 1.0f, N_ATOMS);
  k_deg_scatter<<<cdiv(N_EDGES, T), T, 0, stream>>>(dinv, dst, N_EDGES);
  k_rsqrt_inplace<<<cdiv(N_ATOMS, T), T, 0, stream>>>(dinv, N_ATOMS);

  // GCN layers
  for (int l = 0; l < N_LAYERS; ++l) {
    const float* Wl = W_convs + (size_t)l * HID * HID;
    const float* bl = b_convs + (size_t)l * HID;
    k_gemm_hw<<<(N_ATOMS / 16) / RT, 256, 0, stream>>>(h, Wl, dinv, hw, agg);
    k_edge_scatter<<<cdiv((long long)N_EDGES * (HID / 4), T), T, 0, stream>>>(
        agg, hw, src, dst, dinv, N_EDGES);
    k_bias_relu<<<cdiv((long long)N_ATOMS * (HID / 4), T), T, 0, stream>>>(
        h, agg, bl, N_ATOMS * (HID / 4));
  }

  // bond MLP -> scores, cs
  k_bond_mlp<<<N_BONDS / (16 * NT), 256, 0, stream>>>(
      h, atom_i, atom_j, edge_feat, W1p, b1, W2, b2, scores, cs);

  // segment softmax
  k_seg_init<<<cdiv(N_FRAGS, T), T, 0, stream>>>(segkey, denom, N_FRAGS);
  k_seg_max<<<cdiv(N_BONDS, T), T, 0, stream>>>(segkey, scores, group_id, N_BONDS);
  k_exp<<<cdiv(N_BONDS, T), T, 0, stream>>>(eArr, denom, segkey, scores,
                                            group_id, N_BONDS);
  k_lp<<<cdiv(N_BONDS, T), T, 0, stream>>>(lp, eArr, denom, group_id, N_BONDS);

  // existence propagation (MAX_DEPTH is a module constant == 8)
  k_ex_init<<<cdiv(N_FRAGS, T), T, 0, stream>>>(ex, root_idx, N_FRAGS);
  for (int d = 0; d < MAX_DEPTH; ++d) {
    k_w_compute<<<cdiv(N_BONDS, T), T, 0, stream>>>(wbuf, ex, parent, lp, cs,
                                                    depth, child_b, d, 0, N_BONDS);
    k_w_scatter<<<cdiv(N_BONDS, T), T, 0, stream>>>(ex, child_a, wbuf, depth,
                                                    child_b, d, 0, N_BONDS);
    k_w_compute<<<cdiv(N_BONDS, T), T, 0, stream>>>(wbuf, ex, parent, lp, cs,
                                                    depth, child_b, d, 1, N_BONDS);
    k_w_scatter<<<cdiv(N_BONDS, T), T, 0, stream>>>(ex, child_b, wbuf, depth,
                                                    child_b, d, 1, N_BONDS);
  }

  // spectrum
  k_fill<<<cdiv(MZ, T), T, 0, stream>>>(spec, 0.0f, MZ);
  k_spectrum<<<cdiv(N_BONDS, T), T, 0, stream>>>(spec, ex, child_a, child_b,
                                                 mass_a, mass_b, N_BONDS);
}